// ZigzagReorder_35089882808551
// MI455X (gfx1250) — compile-verified
//
#include <hip/hip_runtime.h>
#include <hip/hip_bf16.h>

// ZigzagReorder: out[b,c,p] = x[b,c, zigzag_idx[p]] for 64x64 planes.
// Pure permutation, HBM-bandwidth bound (~268 MB total -> ~11.5us @ 23.3 TB/s).
// Strategy: one workgroup per (b,c) row (16 KiB contiguous), stage row into LDS
// with gfx1250 async global->LDS b128 copies (ASYNCcnt path), gather from LDS,
// write coalesced float4 stores.

#define ZZ_H 64
#define ZZ_W 64
#define ZZ_N (ZZ_H * ZZ_W)          // 4096 elements per row
#define ZZ_THREADS 256

typedef int v4i __attribute__((vector_size(16)));   // b128 payload type for async builtin

// ---------------------------------------------------------------------------
// Kernel 1: build the zigzag permutation table (length 4096) in d_ws.
// One thread per anti-diagonal s = i + j (127 diagonals). Mirrors reference.
// ---------------------------------------------------------------------------
__global__ void zz_build_idx(int* __restrict__ idx) {
    const int s = threadIdx.x;
    if (s >= ZZ_H + ZZ_W - 1) return;                 // 0..126
    const int lo  = (s > ZZ_W - 1) ? (s - (ZZ_W - 1)) : 0;
    const int hi  = (s < ZZ_H - 1) ? s : (ZZ_H - 1);
    const int len = hi - lo + 1;
    // elements preceding diagonal s
    const int off = (s < ZZ_W)
                        ? (s * (s + 1)) / 2
                        : ZZ_N - ((2 * ZZ_W - 1 - s) * (2 * ZZ_W - s)) / 2;
    for (int t = 0; t < len; ++t) {
        const int i = (s & 1) ? (lo + t) : (hi - t);  // odd: ascend, even: descend
        const int j = s - i;
        idx[off + t] = i * ZZ_W + j;
    }
}

// ---------------------------------------------------------------------------
// Kernel 2: per-row LDS-staged gather.
// ---------------------------------------------------------------------------
__global__ __launch_bounds__(ZZ_THREADS) void zz_gather(
    const float* __restrict__ x, const int* __restrict__ idx,
    float* __restrict__ out) {
    __shared__ float row[ZZ_N];                       // 16 KiB

    const size_t base = (size_t)blockIdx.x * ZZ_N;
    const float* __restrict__ src = x + base;
    float*       __restrict__ dst = out + base;
    const int tid = threadIdx.x;

    // ---- Stage the contiguous 16 KiB row into LDS (coalesced b128) ----
#if __has_builtin(__builtin_amdgcn_global_load_async_to_lds_b128)
    #pragma unroll
    for (int c = 0; c < ZZ_N / (ZZ_THREADS * 4); ++c) {
        const int e = (c * ZZ_THREADS + tid) * 4;     // 16B-aligned
        __builtin_amdgcn_global_load_async_to_lds_b128(
            (__attribute__((address_space(1))) v4i*)(src + e),
            (__attribute__((address_space(3))) v4i*)(row + e),
            /*offset=*/0, /*cpol=*/0);
    }
    // wait for this wave's async copies before the workgroup barrier
  #if __has_builtin(__builtin_amdgcn_s_wait_asynccnt)
    __builtin_amdgcn_s_wait_asynccnt(0);
  #else
    asm volatile("s_wait_asynccnt 0" ::: "memory");
  #endif
#else
    #pragma unroll
    for (int c = 0; c < ZZ_N / (ZZ_THREADS * 4); ++c) {
        const int e = (c * ZZ_THREADS + tid) * 4;
        *(float4*)(&row[e]) = *(const float4*)(src + e);
    }
#endif
    __syncthreads();

    // ---- Gather from LDS, write coalesced float4 stores ----
    #pragma unroll
    for (int c = 0; c < ZZ_N / (ZZ_THREADS * 4); ++c) {
        const int p4 = (c * ZZ_THREADS + tid) * 4;    // 4 consecutive zigzag slots
        const int4 id = *(const int4*)(idx + p4);     // coalesced, L2-hot (16 KiB)
        float4 v;
        v.x = row[id.x];
        v.y = row[id.y];
        v.z = row[id.z];
        v.w = row[id.w];
        *(float4*)(dst + p4) = v;                     // coalesced b128 store
    }
}

extern "C" void kernel_launch(void* const* d_in, const int* in_sizes, int n_in,
                              void* d_out, int out_size, void* d_ws, size_t ws_size,
                              hipStream_t stream) {
    (void)n_in; (void)out_size; (void)ws_size;
    const float* x   = (const float*)d_in[0];
    float*       out = (float*)d_out;
    int*         idx = (int*)d_ws;                    // 4096 ints = 16 KiB scratch

    const int rows = in_sizes[0] / ZZ_N;              // B*C = 8192

    zz_build_idx<<<1, 128, 0, stream>>>(idx);
    zz_gather<<<rows, ZZ_THREADS, 0, stream>>>(x, idx, out);
}